// BlockSparseLinear_64768106824287
// MI455X (gfx1250) — compile-verified
//
#include <hip/hip_runtime.h>

// ---------------------------------------------------------------------------
// Block-sparse linear  y = W_sparse @ x + bias   (all f32)
//   x:             [32768, 1024]   (IN_FEATURES x BATCH)
//   block_weights: [819, 256, 256] (n, out_neuron, in_neuron)
//   bias:          [16384, 1]
//   in_idx/out_idx:[819] int32
//   y:             [16384, 1024]
// ---------------------------------------------------------------------------

#define N_IN_BLOCKS 128
#define N_OUT_BLOCKS 64
#define BIN 256
#define BOUT 256
#define N_BLK 819
#define BATCH 1024
#define OUT_FEATURES (N_OUT_BLOCKS * BOUT)

#define MAXB 128        // CSR capacity per output block (avg occupancy ~12.8)
#define MT 64           // rows of the output block per workgroup
#define NT 256          // batch columns per workgroup
#define KC 32           // K chunk staged through LDS

typedef __attribute__((ext_vector_type(2))) float v2f;
typedef __attribute__((ext_vector_type(8))) float v8f;

// ---------------------------------------------------------------------------
// Kernel 1: bin blocks by output block (tiny; 64 threads scan 819 entries)
// ---------------------------------------------------------------------------
__global__ void bsl_build_csr(const int* __restrict__ out_idx,
                              int* __restrict__ counts,
                              int* __restrict__ lists) {
    int o = threadIdx.x;
    if (o < N_OUT_BLOCKS) {
        int c = 0;
        for (int i = 0; i < N_BLK; ++i) {
            if (out_idx[i] == o) {
                if (c < MAXB) lists[o * MAXB + c] = i;
                ++c;
            }
        }
        counts[o] = (c > MAXB) ? MAXB : c;
    }
}

// ---------------------------------------------------------------------------
// Kernel 2: block-sparse GEMM with f32 WMMA (16x16x4), register accumulation
//   grid = (BATCH/NT, BOUT/MT, N_OUT_BLOCKS), block = 256 threads (8 waves)
//   wave w covers batch columns [w*32, w*32+32) of the WG tile; all waves
//   share the 64-row W slab in LDS.
// ---------------------------------------------------------------------------
__global__ __launch_bounds__(256)
void bsl_gemm(const float* __restrict__ x,
              const float* __restrict__ wblk,
              const float* __restrict__ bias,
              const int* __restrict__ in_idx,
              const int* __restrict__ counts,
              const int* __restrict__ lists,
              float* __restrict__ y) {
    __shared__ float sW[MT * KC];        // 64 x 32  (row-major, stride KC)
    __shared__ float sX[KC * NT];        // 32 x 256 (row-major, stride NT)

    const int tid   = threadIdx.x;
    const int wave  = tid >> 5;          // 0..7
    const int lane  = tid & 31;
    const int ml    = lane & 15;         // M / N position within a 16-tile
    const int khi   = lane >> 4;         // K half select (ISA 16x4 f32 layout)

    const int colb  = blockIdx.x * NT;   // batch column base of WG tile
    const int rowb  = blockIdx.y * MT;   // row base within the output block
    const int oblk  = blockIdx.z;        // output GO-block

    const int cnt = counts[oblk];

    v8f acc[4][2];
    #pragma unroll
    for (int mt = 0; mt < 4; ++mt)
        #pragma unroll
        for (int nt = 0; nt < 2; ++nt)
            acc[mt][nt] = (v8f)0.0f;

    for (int e = 0; e < cnt; ++e) {
        const int n  = lists[oblk * MAXB + e];
        const int ib = in_idx[n];
        const float* __restrict__ wbase =
            wblk + ((size_t)n * BOUT + rowb) * BIN;          // W[n][rowb..][.]
        const float* __restrict__ xbase =
            x + (size_t)ib * BIN * BATCH + colb;             // x rows ib*256..

        for (int k0 = 0; k0 < BIN; k0 += KC) {
            // ---- stage W slab: 64 x 32 floats = 512 float4, 2 per thread
            #pragma unroll
            for (int i = 0; i < 2; ++i) {
                int c   = tid + i * 256;                     // float4 index
                int row = c >> 3;                            // /8 float4 per row
                int q   = c & 7;
                const float4 wv =
                    *(const float4*)(wbase + (size_t)row * BIN + k0 + q * 4);
                *(float4*)(&sW[row * KC + q * 4]) = wv;
            }
            // ---- stage X slab: 32 x 256 floats = 2048 float4, 8 per thread
            #pragma unroll
            for (int i = 0; i < 8; ++i) {
                int c  = tid + i * 256;
                int kk = c >> 6;                             // /64 float4 per row
                int q  = c & 63;
                const float4 xv =
                    *(const float4*)(xbase + (size_t)(k0 + kk) * BATCH + q * 4);
                *(float4*)(&sX[kk * NT + q * 4]) = xv;
            }
            __syncthreads();

            // ---- 8 K-steps of 4, 8 WMMAs each
            #pragma unroll
            for (int kb = 0; kb < KC; kb += 4) {
                v2f afrag[4];
                #pragma unroll
                for (int mt = 0; mt < 4; ++mt) {
                    const int m = mt * 16 + ml;
                    afrag[mt].x = sW[m * KC + kb + 2 * khi + 0];
                    afrag[mt].y = sW[m * KC + kb + 2 * khi + 1];
                }
                v2f bfrag[2];
                #pragma unroll
                for (int nt = 0; nt < 2; ++nt) {
                    const int nn = wave * 32 + nt * 16 + ml;
                    bfrag[nt].x = sX[(kb + 2 * khi + 0) * NT + nn];
                    bfrag[nt].y = sX[(kb + 2 * khi + 1) * NT + nn];
                }
                #pragma unroll
                for (int mt = 0; mt < 4; ++mt)
                    #pragma unroll
                    for (int nt = 0; nt < 2; ++nt)
                        acc[mt][nt] = __builtin_amdgcn_wmma_f32_16x16x4_f32(
                            /*neg_a=*/false, afrag[mt],
                            /*neg_b=*/false, bfrag[nt],
                            /*c_mod=*/(short)0, acc[mt][nt],
                            /*reuse_a=*/false, /*reuse_b=*/false);
            }
            __syncthreads();
        }
    }

    // ---- epilogue: add bias, store. C/D layout: VGPR v holds rows v (lanes
    //      0-15) and v+8 (lanes 16-31); column = lane%16 within each 16-tile.
    const int rhalf = khi * 8;
    #pragma unroll
    for (int mt = 0; mt < 4; ++mt) {
        #pragma unroll
        for (int v = 0; v < 8; ++v) {
            const int row = oblk * BOUT + rowb + mt * 16 + rhalf + v;
            const float bb = bias[row];
            #pragma unroll
            for (int nt = 0; nt < 2; ++nt) {
                const int col = colb + wave * 32 + nt * 16 + ml;
                y[(size_t)row * BATCH + col] = acc[mt][nt][v] + bb;
            }
        }
    }
}

// ---------------------------------------------------------------------------
extern "C" void kernel_launch(void* const* d_in, const int* in_sizes, int n_in,
                              void* d_out, int out_size, void* d_ws, size_t ws_size,
                              hipStream_t stream) {
    const float* x    = (const float*)d_in[0];
    const float* wblk = (const float*)d_in[1];
    const float* bias = (const float*)d_in[2];
    const int* in_idx  = (const int*)d_in[3];
    const int* out_idx = (const int*)d_in[4];
    float* y = (float*)d_out;

    int* counts = (int*)d_ws;                       // 64 ints
    int* lists  = counts + N_OUT_BLOCKS;            // 64*128 ints

    bsl_build_csr<<<1, 64, 0, stream>>>(out_idx, counts, lists);

    dim3 grid(BATCH / NT, BOUT / MT, N_OUT_BLOCKS); // (4, 4, 64)
    bsl_gemm<<<grid, 256, 0, stream>>>(x, wblk, bias, in_idx, counts, lists, y);
}